// GCN_16724602651052
// MI455X (gfx1250) — compile-verified
//
#include <hip/hip_runtime.h>

#define DIN  256
#define DH   128
#define DOUT 32

typedef float v8f __attribute__((ext_vector_type(8)));
typedef float v2f __attribute__((ext_vector_type(2)));

// ---------------- degree / norm ----------------
__global__ void gcn_deg_kernel(const int* __restrict__ dst, float* __restrict__ deg, int E) {
    int i = blockIdx.x * blockDim.x + threadIdx.x;
    if (i < E) atomicAdd(&deg[dst[i]], 1.0f);
}

__global__ void gcn_dis_kernel(float* __restrict__ deg, int N) {
    int i = blockIdx.x * blockDim.x + threadIdx.x;
    if (i < N) deg[i] = rsqrtf(deg[i] + 1.0f);   // +1 self-loop; always > 0
}

// ---------------- GEMM1: h = x @ W1  (f32 WMMA 16x16x4) ----------------
// grid.x = N/16 row tiles, 8 waves/block = 8 column tiles of DH=128
__global__ void gcn_gemm1_kernel(const float* __restrict__ x,
                                 const float* __restrict__ w,
                                 float* __restrict__ h) {
    const int lane = threadIdx.x & 31;
    const int wave = threadIdx.x >> 5;
    const int r    = lane & 15;
    const int half = lane >> 4;          // 0: K pair {0,1}, 1: K pair {2,3}
    const size_t row0 = (size_t)blockIdx.x * 16;
    const int    col0 = wave * 16;

    v8f c = {};
    const float* arow = x + (row0 + r) * DIN + 2 * half;          // A: M=row, K=k+2*half
    const float* bcol = w + (size_t)(2 * half) * DH + col0 + r;   // B: K=k+2*half, N=col

    for (int k = 0; k < DIN; k += 4) {
        v2f a; a.x = arow[k];                 a.y = arow[k + 1];
        v2f b; b.x = bcol[(size_t)k * DH];    b.y = bcol[(size_t)k * DH + DH];
        c = __builtin_amdgcn_wmma_f32_16x16x4_f32(false, a, false, b,
                                                  (short)0, c, false, false);
    }
    float* outp = h + (row0 + 8 * half) * DH + col0 + r;
#pragma unroll
    for (int v = 0; v < 8; ++v) outp[(size_t)v * DH] = c[v];      // M = row0+8*half+v
}

// ---------------- scatter layer1: one wave per edge, 128 feats ----------------
__global__ void gcn_scatter128_kernel(const float* __restrict__ h,
                                      const int* __restrict__ src,
                                      const int* __restrict__ dst,
                                      const float* __restrict__ dis,
                                      float* __restrict__ agg,
                                      int E, int N) {
    int wid  = blockIdx.x * (blockDim.x >> 5) + (threadIdx.x >> 5);
    if (wid >= E + N) return;
    int lane = threadIdx.x & 31;
    int s, d;
    if (wid < E) { s = src[wid]; d = dst[wid]; } else { s = d = wid - E; }
    float w = dis[s] * dis[d];
    const float4* hp = (const float4*)(h + (size_t)s * DH);
    float4 v = hp[lane];                                  // feats 4*lane..4*lane+3
    float* ap = agg + (size_t)d * DH + 4 * lane;
    atomicAdd(ap + 0, v.x * w);
    atomicAdd(ap + 1, v.y * w);
    atomicAdd(ap + 2, v.z * w);
    atomicAdd(ap + 3, v.w * w);
}

// ---------------- GEMM2: h2 = relu(agg1 + b1) @ W2 ----------------
// grid.x = N/16 row tiles, 2 waves/block = 2 column tiles of DOUT=32
__global__ void gcn_gemm2_kernel(const float* __restrict__ g,
                                 const float* __restrict__ b1,
                                 const float* __restrict__ w,
                                 float* __restrict__ h2) {
    const int lane = threadIdx.x & 31;
    const int wave = threadIdx.x >> 5;
    const int r    = lane & 15;
    const int half = lane >> 4;
    const size_t row0 = (size_t)blockIdx.x * 16;
    const int    col0 = wave * 16;

    v8f c = {};
    const float* arow = g  + (row0 + r) * DH + 2 * half;
    const float* bb   = b1 + 2 * half;
    const float* bcol = w  + (size_t)(2 * half) * DOUT + col0 + r;

    for (int k = 0; k < DH; k += 4) {
        v2f a;
        a.x = fmaxf(arow[k]     + bb[k],     0.0f);     // fused bias + ReLU
        a.y = fmaxf(arow[k + 1] + bb[k + 1], 0.0f);
        v2f b; b.x = bcol[(size_t)k * DOUT]; b.y = bcol[(size_t)k * DOUT + DOUT];
        c = __builtin_amdgcn_wmma_f32_16x16x4_f32(false, a, false, b,
                                                  (short)0, c, false, false);
    }
    float* outp = h2 + (row0 + 8 * half) * DOUT + col0 + r;
#pragma unroll
    for (int v = 0; v < 8; ++v) outp[(size_t)v * DOUT] = c[v];
}

// ---------------- scatter layer2: one wave per edge, 32 feats ----------------
__global__ void gcn_scatter32_kernel(const float* __restrict__ h2,
                                     const int* __restrict__ src,
                                     const int* __restrict__ dst,
                                     const float* __restrict__ dis,
                                     float* __restrict__ out,
                                     int E, int N) {
    int wid  = blockIdx.x * (blockDim.x >> 5) + (threadIdx.x >> 5);
    if (wid >= E + N) return;
    int lane = threadIdx.x & 31;
    int s, d;
    if (wid < E) { s = src[wid]; d = dst[wid]; } else { s = d = wid - E; }
    float w = dis[s] * dis[d];
    float v = h2[(size_t)s * DOUT + lane] * w;
    atomicAdd(&out[(size_t)d * DOUT + lane], v);
}

// ---------------- bias + log_softmax, one wave per row (wave32) ----------------
__global__ void gcn_logsoftmax_kernel(float* __restrict__ out,
                                      const float* __restrict__ b2, int N) {
    int row  = blockIdx.x * (blockDim.x >> 5) + (threadIdx.x >> 5);
    if (row >= N) return;
    int lane = threadIdx.x & 31;
    float v = out[(size_t)row * DOUT + lane] + b2[lane];
    float m = v;
#pragma unroll
    for (int o = 16; o > 0; o >>= 1) m = fmaxf(m, __shfl_xor(m, o, 32));
    float e = __expf(v - m);
    float s = e;
#pragma unroll
    for (int o = 16; o > 0; o >>= 1) s += __shfl_xor(s, o, 32);
    out[(size_t)row * DOUT + lane] = v - m - __logf(s);
}

extern "C" void kernel_launch(void* const* d_in, const int* in_sizes, int n_in,
                              void* d_out, int out_size, void* d_ws, size_t ws_size,
                              hipStream_t stream) {
    const float* x   = (const float*)d_in[0];
    const int*   ei  = (const int*)d_in[1];
    const float* W1  = (const float*)d_in[2];
    const float* b1  = (const float*)d_in[3];
    const float* W2  = (const float*)d_in[4];
    const float* b2  = (const float*)d_in[5];
    float*       out = (float*)d_out;

    const int N = in_sizes[0] / DIN;
    const int E = in_sizes[1] / 2;
    const int* src = ei;
    const int* dst = ei + E;

    // workspace layout
    char* ws = (char*)d_ws;
    float* deg  = (float*)ws;                                   // N floats -> dis in place
    size_t off1 = (((size_t)N * 4) + 511) & ~(size_t)511;
    float* h    = (float*)(ws + off1);                          // N*DH floats (reused as h2)
    size_t off2 = off1 + ((((size_t)N * DH * 4) + 511) & ~(size_t)511);
    float* agg1 = (float*)(ws + off2);                          // N*DH floats

    hipMemsetAsync(deg,  0, (size_t)N * 4, stream);
    hipMemsetAsync(agg1, 0, (size_t)N * DH * 4, stream);
    hipMemsetAsync(out,  0, (size_t)N * DOUT * 4, stream);

    gcn_deg_kernel<<<(E + 255) / 256, 256, 0, stream>>>(dst, deg, E);
    gcn_dis_kernel<<<(N + 255) / 256, 256, 0, stream>>>(deg, N);

    gcn_gemm1_kernel<<<N / 16, 256, 0, stream>>>(x, W1, h);

    int nw = E + N;
    gcn_scatter128_kernel<<<(nw + 7) / 8, 256, 0, stream>>>(h, src, dst, deg, agg1, E, N);

    gcn_gemm2_kernel<<<N / 16, 64, 0, stream>>>(agg1, b1, W2, h /* reuse as h2 */);

    gcn_scatter32_kernel<<<(nw + 7) / 8, 256, 0, stream>>>(h, src, dst, deg, out, E, N);

    gcn_logsoftmax_kernel<<<(N + 7) / 8, 256, 0, stream>>>(out, b2, N);
}